// GriffinRecurrence_80642305950357
// MI455X (gfx1250) — compile-verified
//
#include <hip/hip_runtime.h>
#include <math.h>
#include <stdint.h>

// ---------------------------------------------------------------------------
// GriffinRecurrence for MI455X (gfx1250, wave32)
//
//   aiv = x @ W^T  (16384 x 1152, K=2048, fp32)  -> V_WMMA_F32_16X16X4_F32
//   a = sigmoid(a_proj + bias), inp = sqrt(clip(1-a^2)) * sigmoid(i_proj) * v
//   chunked scan (chunk=64) with the reference's clipped log/exp cumsums
//
// K1 fuses GEMM + gates + intra-chunk scan (a 64-row M tile == one chunk),
// with double-buffered async global->LDS panel prefetch (ASYNCcnt path).
// K2 does the 64-chunk inter-chunk scan. K3 adds cross-chunk contribution.
// ---------------------------------------------------------------------------

typedef float v2f __attribute__((ext_vector_type(2)));
typedef float v8f __attribute__((ext_vector_type(8)));

#define D_MODEL 2048
#define D_REC   384
#define M_TOTAL 16384          // 4 * 4096
#define N_CHUNKS_TOTAL 256     // M_TOTAL / 64  (b-major: chunkGlobal = b*64 + c)
#define CHUNK   64
#define KB      64             // K panel width
#define LDA     (KB + 4)       // padded LDS stride (68 floats -> 272B, 16B-aligned rows)
#define PANEL_FLOATS (112 * LDA)   // 64 A-rows + 48 B-rows per panel = 7616 floats
#define EPS_A   1e-8f
#define EPS_LOG 1e-10f

__device__ __forceinline__ float sigmoidf_(float x) {
    return 1.0f / (1.0f + expf(-x));
}

// Async global->LDS copy of 16 bytes per lane (CDNA5 GLOBAL_LOAD_ASYNC_TO_LDS_B128,
// tracked by ASYNCcnt). lds_byte_off is the wave-relative LDS byte address.
__device__ __forceinline__ void async_copy16(uint32_t lds_byte_off, const float* g) {
    asm volatile("global_load_async_to_lds_b128 %0, %1, off"
                 :: "v"(lds_byte_off), "v"(g)
                 : "memory");
}

__device__ __forceinline__ void wait_async0() {
    asm volatile("s_wait_asynccnt 0x0" ::: "memory");
}

__launch_bounds__(128)
__global__ void k1_gemm_gate_intrascan(const float* __restrict__ x,     // [16384][2048]
                                       const float* __restrict__ W,     // [1152][2048]
                                       const float* __restrict__ bias,  // [384]
                                       float* __restrict__ intra_out,   // d_out [16384][384]
                                       float* __restrict__ cum_decay,   // ws    [16384][384]
                                       float* __restrict__ chunk_decay, // ws    [256][384]
                                       float* __restrict__ chunk_state) // ws    [256][384]
{
    // Two K-panels (A: 64xKB, B: 48xKB each), double buffered = 60928 B.
    // The intra-chunk scan staging buffers alias panel 0 (dead after GEMM).
    __shared__ __align__(16) float smem[2 * PANEL_FLOATS];

    const int tileM = blockIdx.x;          // 0..255  == global chunk index
    const int d0    = blockIdx.y * 16;     // channel tile base, 0..368
    const int t     = threadIdx.x;         // 0..127 (4 waves)
    const int wave  = t >> 5;
    const int lane  = t & 31;
    const int r0    = tileM * 64;          // global row (b*4096+s) base

    const uint32_t smem_base = (uint32_t)(uintptr_t)(void*)smem;  // LDS byte offset

    v8f accA = {}; v8f accI = {}; v8f accV = {};

    const int col  = lane & 15;            // N / M-low index within 16
    const int koff = (lane >> 4) * 2;      // K sub-offset per ISA f32 A/B layout
    const int mrow = wave * 16 + col;      // this wave's A row in the 64-row tile

    // Issue one panel's async copies: 8 A-chunks + 6 B-chunks of 16B per thread.
    auto issue_panel = [&](int k0, uint32_t bufbase) {
        #pragma unroll
        for (int it = 0; it < 8; ++it) {
            int idx = t + it * 128;                    // 0..1023
            int row = idx >> 4;
            int qc  = idx & 15;
            async_copy16(bufbase + (uint32_t)(row * LDA + qc * 4) * 4u,
                         x + (size_t)(r0 + row) * D_MODEL + k0 + qc * 4);
        }
        #pragma unroll
        for (int it = 0; it < 6; ++it) {
            int idx = t + it * 128;                    // 0..767
            int row = idx >> 4;                        // 0..47
            int qc  = idx & 15;
            int n   = (row >> 4) * D_REC + d0 + (row & 15);
            async_copy16(bufbase + (uint32_t)((64 + row) * LDA + qc * 4) * 4u,
                         W + (size_t)n * D_MODEL + k0 + qc * 4);
        }
    };

    // Prologue: prefetch panel 0 into buffer 0.
    issue_panel(0, smem_base);
    wait_async0();
    __syncthreads();

    const int NPANELS = D_MODEL / KB;      // 32
    for (int p = 0; p < NPANELS; ++p) {
        // Prefetch next panel into the other buffer while computing this one.
        if (p + 1 < NPANELS)
            issue_panel((p + 1) * KB, smem_base + (uint32_t)(((p + 1) & 1) * PANEL_FLOATS) * 4u);

        const float* As = smem + (p & 1) * PANEL_FLOATS;
        const float* Bs = As + 64 * LDA;

        #pragma unroll
        for (int kk = 0; kk < KB; kk += 4) {
            v2f a  = *(const v2f*)(As + mrow * LDA + kk + koff);
            v2f b0 = *(const v2f*)(Bs + ( 0 + col) * LDA + kk + koff);
            v2f b1 = *(const v2f*)(Bs + (16 + col) * LDA + kk + koff);
            v2f b2 = *(const v2f*)(Bs + (32 + col) * LDA + kk + koff);
            accA = __builtin_amdgcn_wmma_f32_16x16x4_f32(false, a, false, b0, (short)0, accA, false, false);
            accI = __builtin_amdgcn_wmma_f32_16x16x4_f32(false, a, false, b1, (short)0, accI, false, false);
            accV = __builtin_amdgcn_wmma_f32_16x16x4_f32(false, a, false, b2, (short)0, accV, false, false);
        }

        wait_async0();        // next panel landed in LDS
        __syncthreads();      // everyone done reading current panel
    }

    // Scan staging buffers alias panel buffer 0 (all panel reads are behind the
    // final barrier above).
    float* aS = smem;                  // [CHUNK][17]
    float* sS = smem + CHUNK * 17;     // [CHUNK][17]

    // ---- gate epilogue: C layout lane l, vgpr j -> row = j + (l>>4)*8, col = l&15
    const float bb = bias[d0 + col];
    #pragma unroll
    for (int j = 0; j < 8; ++j) {
        float aa  = sigmoidf_(accA[j] + bb);
        float ii  = sigmoidf_(accI[j]);
        float sig = sqrtf(fmaxf(1.0f - aa * aa, EPS_A)) * (ii * accV[j]);
        int rl = wave * 16 + (lane >> 4) * 8 + j;   // row within the 64-row chunk
        aS[rl * 17 + col] = aa;
        sS[rl * 17 + col] = sig;
    }
    __syncthreads();

    // ---- intra-chunk scan (reference numerics, clipped log/exp cumsums) ----
    if (t < 16) {
        const int c = t;                              // channel within tile
        float cum_log = 0.0f, cw = 0.0f, cd = 1.0f, intra = 0.0f;
        for (int step = 0; step < CHUNK; ++step) {
            float aa = aS[step * 17 + c];
            float vv = sS[step * 17 + c];
            cum_log += logf(fmaxf(aa, EPS_LOG));
            cd = expf(cum_log);
            cw += vv / fmaxf(cd, EPS_LOG);
            intra = cd * cw;
            size_t o = (size_t)(r0 + step) * D_REC + d0 + c;
            intra_out[o] = intra;
            cum_decay[o] = cd;
        }
        size_t co = (size_t)tileM * D_REC + d0 + c;
        chunk_decay[co] = cd;       // chunk_total_decay
        chunk_state[co] = intra;    // chunk_final_state
    }
}

// Inter-chunk scan: per (batch, channel), sequential over 64 chunks.
__global__ void k2_interchunk(const float* __restrict__ chunk_decay,
                              const float* __restrict__ chunk_state,
                              float* __restrict__ incoming)   // [256][384]
{
    int idx = blockIdx.x * blockDim.x + threadIdx.x;  // 0..1535
    if (idx >= 4 * D_REC) return;
    int b = idx / D_REC;
    int d = idx % D_REC;
    float cl = 0.0f, cwc = 0.0f;
    float prev_ccd = 1.0f, prev_cwc = 0.0f;
    for (int c = 0; c < 64; ++c) {
        size_t o = (size_t)(b * 64 + c) * D_REC + d;
        incoming[o] = (c == 0) ? 0.0f : prev_ccd * prev_cwc;  // incoming_state
        cl += logf(fmaxf(chunk_decay[o], EPS_LOG));
        float ccd = expf(cl);                                  // cum_chunk_decay
        cwc += chunk_state[o] / fmaxf(ccd, EPS_LOG);           // cum_weighted_chunk
        prev_ccd = ccd; prev_cwc = cwc;
    }
}

// states = intra_state + cum_decay * incoming_state[chunk]
__global__ void k3_combine(float* __restrict__ out,
                           const float* __restrict__ cum_decay,
                           const float* __restrict__ incoming)
{
    size_t idx = (size_t)blockIdx.x * blockDim.x + threadIdx.x;
    if (idx >= (size_t)M_TOTAL * D_REC) return;
    size_t row = idx / D_REC;
    int   d   = (int)(idx % D_REC);
    size_t cg = row >> 6;                  // global chunk index
    out[idx] = out[idx] + cum_decay[idx] * incoming[cg * D_REC + d];
}

extern "C" void kernel_launch(void* const* d_in, const int* in_sizes, int n_in,
                              void* d_out, int out_size, void* d_ws, size_t ws_size,
                              hipStream_t stream) {
    const float* x    = (const float*)d_in[0];   // (4,4096,2048)
    const float* W    = (const float*)d_in[1];   // (1152,2048)
    const float* bias = (const float*)d_in[2];   // (384,)
    float* out = (float*)d_out;                  // (4,4096,384)

    float* ws  = (float*)d_ws;
    float* cd  = ws;                                           // 16384*384
    float* chD = cd  + (size_t)M_TOTAL * D_REC;                // 256*384
    float* chS = chD + (size_t)N_CHUNKS_TOTAL * D_REC;         // 256*384
    float* inc = chS + (size_t)N_CHUNKS_TOTAL * D_REC;         // 256*384

    dim3 g1(M_TOTAL / 64, D_REC / 16);   // 256 x 24
    k1_gemm_gate_intrascan<<<g1, 128, 0, stream>>>(x, W, bias, out, cd, chD, chS);

    k2_interchunk<<<6, 256, 0, stream>>>(chD, chS, inc);

    size_t tot = (size_t)M_TOTAL * D_REC;
    k3_combine<<<(unsigned)((tot + 255) / 256), 256, 0, stream>>>(out, cd, inc);
}